// GCN_78529182040196
// MI455X (gfx1250) — compile-verified
//
#include <hip/hip_runtime.h>
#include <hip/hip_bf16.h>

#define D_FEAT 128

typedef __attribute__((ext_vector_type(2))) float v2f;
typedef __attribute__((ext_vector_type(8))) float v8f;

// Native hardware f32 atomic add (GLOBAL_ATOMIC_ADD_F32), not a CAS loop:
// unsafeAtomicAdd attaches the no-fine-grained/ignore-denormal metadata that
// lets the AMDGPU backend select the native FP atomic.
__device__ __forceinline__ void atomic_add_f32(float* p, float v) {
  unsafeAtomicAdd(p, v);
}

// ---------------- zero fill (float4 granularity) ----------------
__global__ void zero4_kernel(float4* __restrict__ p, long n4) {
  long i = (long)blockIdx.x * blockDim.x + threadIdx.x;
  if (i < n4) p[i] = make_float4(0.f, 0.f, 0.f, 0.f);
}

// ---------------- degree accumulation ----------------
__global__ void degree_kernel(const int* __restrict__ src, const int* __restrict__ dst,
                              float* __restrict__ deg_src, float* __restrict__ deg_dst, int E) {
  int e = blockIdx.x * blockDim.x + threadIdx.x;
  if (e < E) {
    atomic_add_f32(&deg_src[src[e]], 1.0f);
    atomic_add_f32(&deg_dst[dst[e]], 1.0f);
  }
}

// ---------------- deg -> clip(deg,1)^-0.5 in place ----------------
__global__ void norm_kernel(float* __restrict__ d, int n) {
  int i = blockIdx.x * blockDim.x + threadIdx.x;
  if (i < n) d[i] = rsqrtf(fmaxf(d[i], 1.0f));
}

// ---------------- SpMM scatter-add: agg[dst] += h[src] * nsrc[src] ----------------
// One wave32 per edge: 32 lanes x float4 = one 128-float feature row (coalesced).
// Working set (51.2 MB of node features + 51.2 MB accumulator) is L2-resident
// on MI455X's 192 MB L2, so both the gathers and the atomics run at L2 speed.
__global__ void spmm_kernel(const float* __restrict__ h, const int* __restrict__ src,
                            const int* __restrict__ dst, const float* __restrict__ nsrc,
                            float* __restrict__ agg, int E) {
  int wid  = (int)(((long)blockIdx.x * blockDim.x + threadIdx.x) >> 5);
  int lane = threadIdx.x & 31;
  if (wid >= E) return;
  int s = src[wid];
  int d = dst[wid];
  float ns = nsrc[s];
  const float4 v = ((const float4*)(h + (size_t)s * D_FEAT))[lane];
  float* base = agg + (size_t)d * D_FEAT + lane * 4;
  atomic_add_f32(base + 0, v.x * ns);
  atomic_add_f32(base + 1, v.y * ns);
  atomic_add_f32(base + 2, v.z * ns);
  atomic_add_f32(base + 3, v.w * ns);
}

// ---------------- dense GEMM: out = relu?((in .* ndst) @ W + b) ----------------
// in: [N,128], W: [128,DOUT] row-major, out: [N,DOUT].
// Block = one 16-row tile; wave w owns output columns [16w,16w+16) via
// V_WMMA_F32_16X16X4_F32 (full f32 precision), K-loop of 32 steps of 4.
// A-frag layout (16x4 f32): lane l, vgpr j holds A[M=l%16][K=j+2*(l/16)].
// B-frag mirrors with N=l%16. C/D: lane l, vgpr v -> D[v+8*(l/16)][l%16].
template <int DOUT, bool RELU>
__global__ __launch_bounds__(32 * (DOUT / 16)) void gemm_kernel(
    const float* __restrict__ in, const float* __restrict__ ndst,
    const float* __restrict__ W, const float* __restrict__ bias,
    float* __restrict__ out, int nrows) {
  constexpr int WAVES = DOUT / 16;
  __shared__ float As[16 * 132];  // stride 132 avoids LDS bank conflicts

  const int row0 = blockIdx.x * 16;
  const int tid  = threadIdx.x;

  // cooperative load of 16x128 A tile, scaled by norm_dst
  for (int i = tid; i < 16 * D_FEAT; i += 32 * WAVES) {
    int r = i >> 7;
    int c = i & 127;
    int row = row0 + r;
    As[r * 132 + c] = (row < nrows) ? in[(size_t)row * D_FEAT + c] * ndst[row] : 0.0f;
  }
  __syncthreads();

  const int wave = tid >> 5;
  const int lane = tid & 31;
  const int n    = lane & 15;   // output column within tile / A row M
  const int kh   = lane >> 4;   // K half-select (0 or 1)
  const int col  = wave * 16 + n;

  v8f acc = {};
#pragma unroll
  for (int k = 0; k < D_FEAT; k += 4) {
    v2f a, b;
    a[0] = As[n * 132 + k + 2 * kh + 0];
    a[1] = As[n * 132 + k + 2 * kh + 1];
    b[0] = W[(size_t)(k + 2 * kh + 0) * DOUT + col];
    b[1] = W[(size_t)(k + 2 * kh + 1) * DOUT + col];
    acc = __builtin_amdgcn_wmma_f32_16x16x4_f32(
        /*neg_a=*/false, a, /*neg_b=*/false, b,
        /*c_mod=*/(short)0, acc, /*reuse_a=*/false, /*reuse_b=*/false);
  }

  const float bv = bias[col];
#pragma unroll
  for (int v = 0; v < 8; ++v) {
    int m = v + 8 * kh;
    float x = acc[v] + bv;
    if (RELU) x = fmaxf(x, 0.0f);
    if (row0 + m < nrows) out[(size_t)(row0 + m) * DOUT + col] = x;
  }
}

// ---------------- log_softmax over rows of 64 (one wave32 per row) ----------------
__global__ void logsoftmax64_kernel(float* __restrict__ out, int N) {
  int wid  = (int)(((long)blockIdx.x * blockDim.x + threadIdx.x) >> 5);
  int lane = threadIdx.x & 31;
  if (wid >= N) return;
  float* row = out + (size_t)wid * 64;
  float x0 = row[lane];
  float x1 = row[lane + 32];
  float m = fmaxf(x0, x1);
#pragma unroll
  for (int o = 16; o > 0; o >>= 1) m = fmaxf(m, __shfl_xor(m, o, 32));
  float s = expf(x0 - m) + expf(x1 - m);
#pragma unroll
  for (int o = 16; o > 0; o >>= 1) s += __shfl_xor(s, o, 32);
  float ls = m + logf(s);
  row[lane]      = x0 - ls;
  row[lane + 32] = x1 - ls;
}

extern "C" void kernel_launch(void* const* d_in, const int* in_sizes, int n_in,
                              void* d_out, int out_size, void* d_ws, size_t ws_size,
                              hipStream_t stream) {
  const float* features = (const float*)d_in[0];
  const int*   src      = (const int*)d_in[1];
  const int*   dst      = (const int*)d_in[2];
  const float* W1 = (const float*)d_in[3];
  const float* b1 = (const float*)d_in[4];
  const float* W2 = (const float*)d_in[5];
  const float* b2 = (const float*)d_in[6];
  const float* W3 = (const float*)d_in[7];
  const float* b3 = (const float*)d_in[8];
  float* out = (float*)d_out;

  const int N = in_sizes[0] / D_FEAT;
  const int E = in_sizes[1];

  // workspace layout: [norm_src N][norm_dst N][pad][agg N*128][h N*128]
  char* ws = (char*)d_ws;
  float* norm_src = (float*)ws;
  float* norm_dst = norm_src + N;
  size_t normBytes = (((size_t)2 * N * sizeof(float)) + 255) & ~(size_t)255;
  float* agg = (float*)(ws + normBytes);
  float* h   = agg + (size_t)N * D_FEAT;

  const int TB = 256;
  const long degN4    = (long)2 * N / 4;
  const long aggN4    = (long)N * D_FEAT / 4;
  const int  spmmBlks = (E + (TB / 32) - 1) / (TB / 32);
  const int  rowTiles = (N + 15) / 16;

  // degrees -> inverse-sqrt norms
  zero4_kernel<<<(int)((degN4 + TB - 1) / TB), TB, 0, stream>>>((float4*)norm_src, degN4);
  degree_kernel<<<(E + TB - 1) / TB, TB, 0, stream>>>(src, dst, norm_src, norm_dst, E);
  norm_kernel<<<(2 * N + TB - 1) / TB, TB, 0, stream>>>(norm_src, 2 * N);

  // layer 1: features -> h
  zero4_kernel<<<(int)((aggN4 + TB - 1) / TB), TB, 0, stream>>>((float4*)agg, aggN4);
  spmm_kernel<<<spmmBlks, TB, 0, stream>>>(features, src, dst, norm_src, agg, E);
  gemm_kernel<128, true><<<rowTiles, 256, 0, stream>>>(agg, norm_dst, W1, b1, h, N);

  // layer 2: h -> h
  zero4_kernel<<<(int)((aggN4 + TB - 1) / TB), TB, 0, stream>>>((float4*)agg, aggN4);
  spmm_kernel<<<spmmBlks, TB, 0, stream>>>(h, src, dst, norm_src, agg, E);
  gemm_kernel<128, true><<<rowTiles, 256, 0, stream>>>(agg, norm_dst, W2, b2, h, N);

  // layer 3: h -> out (no relu), then log_softmax in place
  zero4_kernel<<<(int)((aggN4 + TB - 1) / TB), TB, 0, stream>>>((float4*)agg, aggN4);
  spmm_kernel<<<spmmBlks, TB, 0, stream>>>(h, src, dst, norm_src, agg, E);
  gemm_kernel<64, false><<<rowTiles, 128, 0, stream>>>(agg, norm_dst, W3, b3, out, N);
  logsoftmax64_kernel<<<((long)N * 32 + TB - 1) / TB, TB, 0, stream>>>(out, N);
}